// Generator_79577154060611
// MI455X (gfx1250) — compile-verified
//
#include <hip/hip_runtime.h>

// ---------------- Problem constants (match reference) ----------------
#define N_DEG 4
#define RANK  128
#define H_    256
#define W_    256
#define S_    (H_ * W_)       // 65536
#define B_    32
#define EPS_  1e-5f

#define KCHUNK  256           // K elements per workgroup in split-K GEMM
#define NK_WGS  (S_ / KCHUNK) // 256 workgroups

// ---------------- Workspace layout (floats) ----------------
// [0] sum, [1] sumsq, [2] scale, [3] shift
// [WS_T .. WS_T + 4*32*128)  : t accumulators (split-K targets)
// [WS_H .. WS_H + 32*128)    : final h
#define WS_T      4
#define WS_H      (WS_T + N_DEG * B_ * RANK)
#define WS_TOTAL  (WS_H + B_ * RANK)

typedef float v2f __attribute__((ext_vector_type(2)));
typedef float v8f __attribute__((ext_vector_type(8)));

// ---------------- Kernel 0: zero workspace accumulators ----------------
__global__ void k_zero(float* __restrict__ ws, int n) {
    int i = blockIdx.x * blockDim.x + threadIdx.x;
    if (i < n) ws[i] = 0.0f;
}

// ---------------- Kernel 1: batch-norm stats (sum, sumsq) ----------------
__global__ __launch_bounds__(256) void k_stats(const float* __restrict__ x,
                                               float* __restrict__ ws) {
    int tid    = blockIdx.x * blockDim.x + threadIdx.x;
    int stride = gridDim.x * blockDim.x;
    float s = 0.0f, ss = 0.0f;
    for (int i = tid; i < B_ * S_; i += stride) {
        float v = x[i];
        s  += v;
        ss += v * v;
    }
    // wave32 reduction
    #pragma unroll
    for (int off = 16; off > 0; off >>= 1) {
        s  += __shfl_down(s,  off, 32);
        ss += __shfl_down(ss, off, 32);
    }
    if ((threadIdx.x & 31) == 0) {
        atomicAdd(&ws[0], s);
        atomicAdd(&ws[1], ss);
    }
}

// ---------------- Kernel 2: finalize BN scalars ----------------
__global__ void k_scalars(float* __restrict__ ws,
                          const float* __restrict__ gamma,
                          const float* __restrict__ bn_bias) {
    const float inv_n = 1.0f / (float)(B_ * S_);
    float mean = ws[0] * inv_n;
    float var  = ws[1] * inv_n - mean * mean;
    float rstd = rsqrtf(var + EPS_);
    float scale = gamma[0] * rstd;
    float shift = bn_bias[0] - mean * scale;
    ws[2] = scale;
    ws[3] = shift;
}

// ---------------- Kernel 3: split-K WMMA GEMM t[n] = z @ U[n] ----------------
// Grid: NK_WGS blocks x 256 threads (8 waves). Each wave owns one N-tile (16 of 128).
// A-fragment (z, 16x4 f32): lanes 0-15 rows M, v0=K+0 v1=K+1 ; lanes 16-31 v0=K+2 v1=K+3
// B-fragment (U, 4x16 f32): VGPR0 -> K (lo half) / K+2 (hi half), VGPR1 -> K+1 / K+3
__global__ __launch_bounds__(256) void k_poly_gemm(const float* __restrict__ x,
                                                   const float* __restrict__ U,
                                                   float* __restrict__ ws) {
    __shared__ float zs[KCHUNK * 33];   // padded: row stride 33 words, bank-conflict free

    const float scale = ws[2];
    const float shift = ws[3];
    const int   k0    = blockIdx.x * KCHUNK;

    // Stage normalized z chunk: zs[kl*33 + b] = z[b, k0+kl]
    for (int idx = threadIdx.x; idx < B_ * KCHUNK; idx += 256) {
        int b  = idx >> 8;               // KCHUNK == 256
        int kl = idx & (KCHUNK - 1);
        zs[kl * 33 + b] = x[(size_t)b * S_ + k0 + kl] * scale + shift;
    }
    __syncthreads();

    const int lane = threadIdx.x & 31;
    const int wid  = threadIdx.x >> 5;       // N-tile 0..7
    const int ln   = lane & 15;
    const int kb   = (lane < 16) ? 0 : 2;    // K sub-offset for this half-wave
    const int ncol = wid * 16 + ln;

    v8f acc[N_DEG][2];
    #pragma unroll
    for (int n = 0; n < N_DEG; ++n) {
        acc[n][0] = (v8f){};
        acc[n][1] = (v8f){};
    }

    const float* Up[N_DEG];
    #pragma unroll
    for (int n = 0; n < N_DEG; ++n)
        Up[n] = U + ((size_t)n * S_ + k0) * RANK + ncol;

    for (int kk = 0; kk < KCHUNK; kk += 4) {
        v2f a0, a1;
        a0.x = zs[(kk + kb    ) * 33 + ln];
        a0.y = zs[(kk + kb + 1) * 33 + ln];
        a1.x = zs[(kk + kb    ) * 33 + 16 + ln];
        a1.y = zs[(kk + kb + 1) * 33 + 16 + ln];

        #pragma unroll
        for (int n = 0; n < N_DEG; ++n) {
            const float* up = Up[n] + (size_t)kk * RANK;
            v2f bf;
            bf.x = up[(size_t)(kb    ) * RANK];
            bf.y = up[(size_t)(kb + 1) * RANK];
            acc[n][0] = __builtin_amdgcn_wmma_f32_16x16x4_f32(
                false, a0, false, bf, (short)0, acc[n][0], false, false);
            acc[n][1] = __builtin_amdgcn_wmma_f32_16x16x4_f32(
                false, a1, false, bf, (short)0, acc[n][1], false, false);
        }
    }

    // Split-K reduction into global accumulator via f32 atomics.
    float* t = ws + WS_T;
    const int mrow = (lane < 16) ? 0 : 8;
    #pragma unroll
    for (int n = 0; n < N_DEG; ++n) {
        #pragma unroll
        for (int m = 0; m < 2; ++m) {
            #pragma unroll
            for (int i = 0; i < 8; ++i) {
                int brow = m * 16 + i + mrow;
                atomicAdd(&t[(size_t)n * (B_ * RANK) + brow * RANK + ncol],
                          acc[n][m][i]);
            }
        }
    }
}

// ---------------- Kernel 4: CCP recurrence h = t0; h = tn*h + h ----------------
__global__ void k_hfin(float* __restrict__ ws) {
    int idx = blockIdx.x * blockDim.x + threadIdx.x;
    if (idx >= B_ * RANK) return;
    const float* t = ws + WS_T;
    float h = t[idx];
    #pragma unroll
    for (int n = 1; n < N_DEG; ++n) {
        float tn = t[(size_t)n * (B_ * RANK) + idx];
        h = fmaf(tn, h, h);
    }
    ws[WS_H + idx] = h;
}

// ---------------- Kernel 5: output WMMA GEMM out = h @ C^T + beta ----------------
// Grid: 512 blocks x 256 threads; wave wid handles N-tile = blockIdx*8 + wid (of 4096).
__global__ __launch_bounds__(256) void k_out_gemm(const float* __restrict__ Cm,
                                                  const float* __restrict__ beta,
                                                  const float* __restrict__ ws,
                                                  float* __restrict__ out) {
    __shared__ float hs[B_ * 129];   // padded row stride 129

    const float* h = ws + WS_H;
    for (int idx = threadIdx.x; idx < B_ * RANK; idx += 256) {
        int b = idx >> 7;
        int r = idx & (RANK - 1);
        hs[b * 129 + r] = h[idx];
    }
    __syncthreads();

    const int lane  = threadIdx.x & 31;
    const int wid   = threadIdx.x >> 5;
    const int ln    = lane & 15;
    const int kb    = (lane < 16) ? 0 : 2;
    const int ntile = blockIdx.x * 8 + wid;
    const int scol  = ntile * 16 + ln;

    const float* crow = Cm + (size_t)scol * RANK;   // C[scol][:]

    v8f acc0 = (v8f){};
    v8f acc1 = (v8f){};

    for (int kk = 0; kk < RANK; kk += 4) {
        v2f a0, a1, bf;
        a0.x = hs[(ln     ) * 129 + kk + kb];
        a0.y = hs[(ln     ) * 129 + kk + kb + 1];
        a1.x = hs[(16 + ln) * 129 + kk + kb];
        a1.y = hs[(16 + ln) * 129 + kk + kb + 1];
        bf   = *(const v2f*)(crow + kk + kb);        // contiguous K,K+1 pair
        acc0 = __builtin_amdgcn_wmma_f32_16x16x4_f32(
            false, a0, false, bf, (short)0, acc0, false, false);
        acc1 = __builtin_amdgcn_wmma_f32_16x16x4_f32(
            false, a1, false, bf, (short)0, acc1, false, false);
    }

    const float bv   = beta[scol];
    const int   mrow = (lane < 16) ? 0 : 8;
    #pragma unroll
    for (int i = 0; i < 8; ++i) {
        out[(size_t)(     i + mrow) * S_ + scol] = acc0[i] + bv;
        out[(size_t)(16 + i + mrow) * S_ + scol] = acc1[i] + bv;
    }
}

// ---------------- Host launcher ----------------
extern "C" void kernel_launch(void* const* d_in, const int* in_sizes, int n_in,
                              void* d_out, int out_size, void* d_ws, size_t ws_size,
                              hipStream_t stream) {
    const float* x       = (const float*)d_in[0];   // [32,1,256,256]
    const float* U       = (const float*)d_in[1];   // [4,65536,128]
    const float* Cm      = (const float*)d_in[2];   // [65536,128]
    const float* beta    = (const float*)d_in[3];   // [65536]
    const float* gamma   = (const float*)d_in[4];   // [1]
    const float* bn_bias = (const float*)d_in[5];   // [1]
    float*       out     = (float*)d_out;           // [32,1,256,256]
    float*       ws      = (float*)d_ws;

    (void)in_sizes; (void)n_in; (void)out_size; (void)ws_size;

    k_zero<<<(WS_TOTAL + 255) / 256, 256, 0, stream>>>(ws, WS_TOTAL);
    k_stats<<<256, 256, 0, stream>>>(x, ws);
    k_scalars<<<1, 1, 0, stream>>>(ws, gamma, bn_bias);
    k_poly_gemm<<<NK_WGS, 256, 0, stream>>>(x, U, ws);
    k_hfin<<<(B_ * RANK + 255) / 256, 256, 0, stream>>>(ws);
    k_out_gemm<<<(S_ / 16) / 8, 256, 0, stream>>>(Cm, beta, ws, out);
}